// RNNLM_30262339568168
// MI455X (gfx1250) — compile-verified
//
#include <hip/hip_runtime.h>

#define VOCAB 32000
#define EMB   32
#define HID   16
#define SEQ   128
#define BATCH 32
#define NTILES (VOCAB / 16)       // 2000
#define ROWS   (SEQ * BATCH)      // 4096
#define TPW    (NTILES / 8)       // 250 tiles per wave (contiguous)

typedef float v2f __attribute__((ext_vector_type(2)));
typedef float v8f __attribute__((ext_vector_type(8)));

// ---------------------------------------------------------------------------
// Kernel A: serial recurrence  h_t = tanh(W_ih @ [x_t ; h_{t-1}] + b_ih)
// One workgroup, 512 threads = (batch b = tid>>4, hidden j = tid&15).
// Writes h_seq[t][b][j] to workspace.
// ---------------------------------------------------------------------------
__global__ __launch_bounds__(512) void rnn_cell_kernel(
    const int*   __restrict__ input,   // [SEQ][BATCH]
    const float* __restrict__ h0,      // [BATCH][HID]
    const float* __restrict__ emb,     // [VOCAB][EMB]
    const float* __restrict__ W_ih,    // [HID][EMB+HID]
    const float* __restrict__ b_ih,    // [HID]
    float*       __restrict__ h_seq)   // [SEQ][BATCH][HID]
{
    __shared__ float sW[HID][EMB + HID];
    __shared__ float sb[HID];
    __shared__ float sx[BATCH][EMB];
    __shared__ float sh[2][BATCH][HID];

    const int tid = threadIdx.x;

    for (int i = tid; i < HID * (EMB + HID); i += 512)
        sW[i / (EMB + HID)][i % (EMB + HID)] = W_ih[i];
    if (tid < HID) sb[tid] = b_ih[tid];
    if (tid < BATCH * HID) sh[0][tid / HID][tid % HID] = h0[tid];
    __syncthreads();

    const int b = tid >> 4;
    const int j = tid & 15;
    int cur = 0;

    for (int t = 0; t < SEQ; ++t) {
        // cooperative embedding gather: 32 samples x 32 floats
        for (int i = tid; i < BATCH * EMB; i += 512) {
            int bb = i >> 5, e = i & 31;
            sx[bb][e] = emb[(long)input[t * BATCH + bb] * EMB + e];
        }
        __syncthreads();

        float pre = sb[j];
        #pragma unroll
        for (int e = 0; e < EMB; ++e) pre += sW[j][e] * sx[b][e];
        #pragma unroll
        for (int k = 0; k < HID; ++k) pre += sW[j][EMB + k] * sh[cur][b][k];
        float v = tanhf(pre);

        sh[cur ^ 1][b][j] = v;                    // double buffer: no WAR on sh[cur]
        h_seq[(t * BATCH + b) * HID + j] = v;
        cur ^= 1;
        __syncthreads();
    }
}

// ---------------------------------------------------------------------------
// Kernel B: logits = h_seq @ W_ho.T + b_ho, fused log_softmax over VOCAB.
// Block = 16 rows (M), 8 waves; wave w owns the contiguous tile range
// [w*250, (w+1)*250) so its 64B store segments stream sequentially per row.
// 3 passes (max, sum-exp, write), logits recomputed via WMMA each pass
// (W_ho = 2MB stays in L2; recompute beats spilling 524MB of raw logits).
// ---------------------------------------------------------------------------
__global__ __launch_bounds__(256) void logits_softmax_kernel(
    const float* __restrict__ h_seq,   // [ROWS][HID]
    const float* __restrict__ W_ho,    // [VOCAB][HID]
    const float* __restrict__ b_ho,    // [VOCAB]
    float*       __restrict__ out)     // [ROWS][VOCAB]
{
    __shared__ float s_red[8][16];
    __shared__ float s_rowmax[16];
    __shared__ float s_corr[16];

    const int tid  = threadIdx.x;
    const int wave = tid >> 5;
    const int lane = tid & 31;
    const int half = lane >> 4;   // 0: rows i, 1: rows i+8 (C/D layout)
    const int l15  = lane & 15;   // N within tile
    const int row0 = blockIdx.x * 16;
    const int nt0  = wave * TPW;
    const int nt1  = nt0 + TPW;

    // A fragments: 16x16 (M x K) as four 16x4 chunks.
    // f32 A layout: VGPR0 = K(4c+0) | K(4c+2), VGPR1 = K(4c+1) | K(4c+3)
    // -> float2 load at k = 4c + 2*half from row-major h_seq.
    v2f a[4];
    #pragma unroll
    for (int c = 0; c < 4; ++c) {
        int k = 4 * c + 2 * half;
        a[c] = *(const v2f*)&h_seq[(row0 + l15) * HID + k];
    }

    auto compute_tile = [&](int n0) -> v8f {
        float bias = b_ho[n0 + l15];   // bias depends only on column N
        v8f acc;
        #pragma unroll
        for (int i = 0; i < 8; ++i) acc[i] = bias;
        #pragma unroll
        for (int c = 0; c < 4; ++c) {
            int k = 4 * c + 2 * half;
            v2f bm = *(const v2f*)&W_ho[(long)(n0 + l15) * HID + k]; // B[k][n]=W_ho[n][k]
            acc = __builtin_amdgcn_wmma_f32_16x16x4_f32(
                false, a[c], false, bm, (short)0, acc, false, false);
        }
        return acc;
    };

    // ---- Pass 1: row max -------------------------------------------------
    float rm[8];
    #pragma unroll
    for (int i = 0; i < 8; ++i) rm[i] = -3.4e38f;
    for (int nt = nt0; nt < nt1; ++nt) {
        v8f acc = compute_tile(nt * 16);
        #pragma unroll
        for (int i = 0; i < 8; ++i) rm[i] = fmaxf(rm[i], acc[i]);
    }
    #pragma unroll
    for (int i = 0; i < 8; ++i)
        #pragma unroll
        for (int m = 1; m < 16; m <<= 1)
            rm[i] = fmaxf(rm[i], __shfl_xor(rm[i], m, 32));
    if (l15 == 0) {
        #pragma unroll
        for (int i = 0; i < 8; ++i) s_red[wave][i + 8 * half] = rm[i];
    }
    __syncthreads();
    if (tid < 16) {
        float m = s_red[0][tid];
        #pragma unroll
        for (int w = 1; w < 8; ++w) m = fmaxf(m, s_red[w][tid]);
        s_rowmax[tid] = m;
    }
    __syncthreads();
    float mrow[8];
    #pragma unroll
    for (int i = 0; i < 8; ++i) mrow[i] = s_rowmax[i + 8 * half];

    // ---- Pass 2: sum of exp, correction = max + log(sum) -----------------
    float rs[8];
    #pragma unroll
    for (int i = 0; i < 8; ++i) rs[i] = 0.0f;
    for (int nt = nt0; nt < nt1; ++nt) {
        v8f acc = compute_tile(nt * 16);
        #pragma unroll
        for (int i = 0; i < 8; ++i) rs[i] += __expf(acc[i] - mrow[i]);
    }
    #pragma unroll
    for (int i = 0; i < 8; ++i)
        #pragma unroll
        for (int m = 1; m < 16; m <<= 1)
            rs[i] += __shfl_xor(rs[i], m, 32);
    if (l15 == 0) {
        #pragma unroll
        for (int i = 0; i < 8; ++i) s_red[wave][i + 8 * half] = rs[i];
    }
    __syncthreads();
    if (tid < 16) {
        float s = 0.0f;
        #pragma unroll
        for (int w = 0; w < 8; ++w) s += s_red[w][tid];
        s_corr[tid] = s_rowmax[tid] + __logf(s);
    }
    __syncthreads();
    float crow[8];
    #pragma unroll
    for (int i = 0; i < 8; ++i) crow[i] = s_corr[i + 8 * half];

    // ---- Pass 3: recompute and store logit - corr ------------------------
    for (int nt = nt0; nt < nt1; ++nt) {
        int n0 = nt * 16;
        v8f acc = compute_tile(n0);
        #pragma unroll
        for (int i = 0; i < 8; ++i) {
            long r = row0 + i + 8 * half;
            out[r * VOCAB + n0 + l15] = acc[i] - crow[i];
        }
    }
}

extern "C" void kernel_launch(void* const* d_in, const int* in_sizes, int n_in,
                              void* d_out, int out_size, void* d_ws, size_t ws_size,
                              hipStream_t stream) {
    const int*   input = (const int*)d_in[0];
    const float* h0    = (const float*)d_in[1];
    const float* emb   = (const float*)d_in[2];
    const float* W_ih  = (const float*)d_in[3];
    const float* b_ih  = (const float*)d_in[4];
    const float* W_ho  = (const float*)d_in[5];
    const float* b_ho  = (const float*)d_in[6];
    float* out   = (float*)d_out;
    float* h_seq = (float*)d_ws;   // 128*32*16 f32 = 256 KB scratch

    rnn_cell_kernel<<<1, 512, 0, stream>>>(input, h0, emb, W_ih, b_ih, h_seq);
    logits_softmax_kernel<<<ROWS / 16, 256, 0, stream>>>(h_seq, W_ho, b_ho, out);
}